// NTM_13237089206958
// MI455X (gfx1250) — compile-verified
//
#include <hip/hip_runtime.h>
#include <cstddef>

// NTM cell for MI455X (gfx1250). Bandwidth-bound on the 128MB `memory`
// tensor (fits in 192MB L2 after first touch). Dense GEMMs use
// v_wmma_f32_16x16x4_f32 (wave32, one 16x16 tile per wave); the dominant
// memory-scan kernel stages rows into LDS with async global->LDS copies.

constexpr int B_   = 64;
constexpr int IN_  = 256;
constexpr int H_   = 512;
constexpr int M_   = 64;
constexpr int N_   = 8192;
constexpr int G4H_ = 2048;

typedef __attribute__((ext_vector_type(2))) float v2f;
typedef __attribute__((ext_vector_type(8))) float v8f;
typedef int v4i_gcc __attribute__((vector_size(16)));   // matches builtin's pointee

// ---- workspace layout (float offsets) ----
constexpr size_t WS_KR    = 0;                       // 64x64  read key
constexpr size_t WS_KW    = 4096;                    // 64x64  write key
constexpr size_t WS_CTLR  = 8192;                    // beta,g,gamma,nk,s (512)
constexpr size_t WS_CTLW  = 8704;
constexpr size_t WS_SCORE = 9216;                    // B*N scores
constexpr size_t WS_X     = WS_SCORE + (size_t)B_*N_;        // B x 320
constexpr size_t WS_GATES = WS_X + (size_t)B_*(IN_+M_);      // B x 2048
constexpr size_t WS_WALL  = WS_GATES + (size_t)B_*G4H_;      // B x 128

// ------------------------------------------------------------------
// Async global->LDS copy of 16 bytes (GLOBAL_LOAD_ASYNC_TO_LDS_B128),
// guarded so the file still compiles if the builtin is unavailable.
// Builtin signature (from probe): (int4 AS1*, int4 AS3*, imm offset, imm cpol)
// with int4 = __attribute__((vector_size(16))) int.
// ------------------------------------------------------------------
__device__ __forceinline__ void async_copy16(const float* g, float* l)
{
#if __has_builtin(__builtin_amdgcn_global_load_async_to_lds_b128)
    __builtin_amdgcn_global_load_async_to_lds_b128(
        (__attribute__((address_space(1))) v4i_gcc*)g,
        (__attribute__((address_space(3))) v4i_gcc*)l,
        0, 0);
#else
    *(float4*)l = *(const float4*)g;     // synchronous fallback
#endif
}

__device__ __forceinline__ void wait_async0()
{
#if __has_builtin(__builtin_amdgcn_s_wait_asynccnt)
    __builtin_amdgcn_s_wait_asynccnt(0);
#else
    asm volatile("s_wait_asynccnt 0" ::: "memory");
#endif
}

// ------------------------------------------------------------------
// Generic fp32 WMMA GEMM: out = act( A1(MxK1) @ W1(NxK1)^T + bias1
//                                  [+ A2(MxK2) @ W2(NxK2)^T + bias2] )
// One wave32 per 16x16 output tile; K stepped by 4 per v_wmma_f32_16x16x4_f32.
// A-frag layout (ISA 7.12.2, 32-bit A 16x4): lanes 0-15 K={k0,k0+1},
// lanes 16-31 K={k0+2,k0+3}; C/D: VGPR v -> row v (+8 for hi lanes).
// ------------------------------------------------------------------
__global__ __launch_bounds__(32)
void wmma_gemm_kernel(const float* __restrict__ A1, const float* __restrict__ W1,
                      const float* __restrict__ bias1, int K1,
                      const float* __restrict__ A2, const float* __restrict__ W2,
                      const float* __restrict__ bias2, int K2,
                      float* __restrict__ out, int Ncols, int act)
{
    const int lane = threadIdx.x;
    const int ntiles = Ncols >> 4;
    const int mt = blockIdx.x / ntiles;
    const int nt = blockIdx.x % ntiles;
    const bool hi = lane >= 16;
    const int r = lane & 15;
    const int arow = mt * 16 + r;
    const int bcol = nt * 16 + r;

    v8f c = {0.f,0.f,0.f,0.f,0.f,0.f,0.f,0.f};

    for (int k0 = 0; k0 < K1; k0 += 4) {
        const int ka = k0 + (hi ? 2 : 0);
        v2f a, b;
        a.x = A1[(size_t)arow * K1 + ka];
        a.y = A1[(size_t)arow * K1 + ka + 1];
        b.x = W1[(size_t)bcol * K1 + ka];
        b.y = W1[(size_t)bcol * K1 + ka + 1];
        c = __builtin_amdgcn_wmma_f32_16x16x4_f32(false, a, false, b,
                                                  (short)0, c, false, false);
    }
    if (A2) {
        for (int k0 = 0; k0 < K2; k0 += 4) {
            const int ka = k0 + (hi ? 2 : 0);
            v2f a, b;
            a.x = A2[(size_t)arow * K2 + ka];
            a.y = A2[(size_t)arow * K2 + ka + 1];
            b.x = W2[(size_t)bcol * K2 + ka];
            b.y = W2[(size_t)bcol * K2 + ka + 1];
            c = __builtin_amdgcn_wmma_f32_16x16x4_f32(false, a, false, b,
                                                      (short)0, c, false, false);
        }
    }

    float bias = bias1[bcol] + (bias2 ? bias2[bcol] : 0.0f);
    const int rbase = mt * 16 + (hi ? 8 : 0);
#pragma unroll
    for (int v = 0; v < 8; ++v) {
        float val = c[v] + bias;
        if (act == 1) val = tanhf(val);
        out[(size_t)(rbase + v) * Ncols + bcol] = val;
    }
}

// ------------------------------------------------------------------
// Controller scalars: cv = h@cw^T+cb (beta/g/gamma), s = softmax(h@sw^T+sb),
// nk = sum(k^2).  1 block x 64 threads (thread = batch row); tiny work.
// ctl layout: [0:64) beta  [64:128) g  [128:192) gamma  [192:256) nk
//             [256:256+3B) s
// ------------------------------------------------------------------
__global__ __launch_bounds__(64)
void ctl_kernel(const float* __restrict__ hsrc,
                const float* __restrict__ cw, const float* __restrict__ cb,
                const float* __restrict__ sw, const float* __restrict__ sb,
                const float* __restrict__ k,  float* __restrict__ ctl)
{
    const int b = threadIdx.x;
    float cv[3], sv[3];
    const float* hr = hsrc + (size_t)b * H_;
#pragma unroll
    for (int j = 0; j < 3; ++j) {
        float dc = 0.f, ds = 0.f;
        const float* cwr = cw + (size_t)j * H_;
        const float* swr = sw + (size_t)j * H_;
        for (int t = 0; t < H_; ++t) { float hv = hr[t]; dc += hv*cwr[t]; ds += hv*swr[t]; }
        cv[j] = dc + cb[j];
        sv[j] = ds + sb[j];
    }
    float mx = fmaxf(sv[0], fmaxf(sv[1], sv[2]));
    float e0 = expf(sv[0]-mx), e1 = expf(sv[1]-mx), e2 = expf(sv[2]-mx);
    float esum = e0 + e1 + e2;

    float nk = 0.f;
    for (int m = 0; m < M_; ++m) { float kv = k[b*M_ + m]; nk += kv*kv; }

    ctl[b]        = fmaxf(cv[0], 0.f) + 1e-4f;            // beta
    ctl[64 + b]   = 1.f / (1.f + expf(-cv[1]));           // g
    ctl[128 + b]  = fmaxf(cv[2], 0.f) + 1.0001f;          // gamma
    ctl[192 + b]  = nk;
    ctl[256 + b*3 + 0] = e0 / esum;
    ctl[256 + b*3 + 1] = e1 / esum;
    ctl[256 + b*3 + 2] = e2 / esum;
}

// ------------------------------------------------------------------
// Score pass: streams full 128MB memory once.
// score[b,n] = beta * <k[b], Mem[b,n,:]> / (||Mem[b,n,:]||^2 * nk[b])
// 256 threads handle 256 rows per block in 4 chunks of 64 rows.
// Each chunk (16KB, contiguous in global memory) is staged into LDS with
// async b128 copies: consecutive lanes move consecutive 16B chunks, so the
// global side is perfectly linear. LDS rows are padded to 68 floats so the
// b128 reads in the compute phase are bank-conflict-free. 4 lanes per row;
// partials combined with wave32 shuffles.
// ------------------------------------------------------------------
__global__ __launch_bounds__(256)
void score_kernel(const float* __restrict__ mem, const float* __restrict__ k,
                  const float* __restrict__ ctl, float* __restrict__ score)
{
    constexpr int ROWS = 64;                  // rows per chunk
    constexpr int RS   = 68;                  // padded row stride (floats)
    __shared__ float tile[ROWS * RS];
    __shared__ float kb[M_];

    const int tid = threadIdx.x;
    const int b   = blockIdx.x >> 5;          // N/256 = 32 blocks per b
    const int n0  = (blockIdx.x & 31) << 8;   // 256 rows per block
    if (tid < M_) kb[tid] = k[b * M_ + tid];

    const float beta = ctl[b];
    const float nk   = ctl[192 + b];
    const int r    = tid >> 2;                // row in chunk (0..63)
    const int part = tid & 3;                 // quarter of the row

    for (int ch = 0; ch < 4; ++ch) {
        const float* gbase = mem + ((size_t)b * N_ + n0 + ch * ROWS) * M_;
#pragma unroll
        for (int i = 0; i < 4; ++i) {
            const int cidx = tid + i * 256;   // 16B-chunk index in the tile
            const int rr = cidx >> 4, q = cidx & 15;
            async_copy16(gbase + (size_t)cidx * 4, &tile[rr * RS + q * 4]);
        }
        wait_async0();
        __syncthreads();

        float dot = 0.f, nm = 0.f;
#pragma unroll
        for (int j = 0; j < 4; ++j) {
            const float4 v = *(const float4*)&tile[r * RS + part * 16 + 4 * j];
            const int kbase = part * 16 + 4 * j;
            dot += kb[kbase+0]*v.x + kb[kbase+1]*v.y + kb[kbase+2]*v.z + kb[kbase+3]*v.w;
            nm  += v.x*v.x + v.y*v.y + v.z*v.z + v.w*v.w;
        }
        dot += __shfl_xor(dot, 1, 32);  dot += __shfl_xor(dot, 2, 32);
        nm  += __shfl_xor(nm,  1, 32);  nm  += __shfl_xor(nm,  2, 32);
        if (part == 0)
            score[(size_t)b * N_ + n0 + ch * ROWS + r] = beta * dot / (nm * nk);
        __syncthreads();
    }
}

// ------------------------------------------------------------------
// Addressing: softmax over N=8192 (staged in 32KB LDS), interpolate with
// old head, circular SR=3 shift (N is pow2 -> mask wrap), sharpen, renorm.
// One workgroup per batch row.
// ------------------------------------------------------------------
__global__ __launch_bounds__(256)
void addr_kernel(const float* __restrict__ score, const float* __restrict__ ctl,
                 const float* __restrict__ head_tm1, float* __restrict__ head_out)
{
    __shared__ float buf[N_];
    __shared__ float red[256];
    const int b = blockIdx.x;
    const int tid = threadIdx.x;
    const float* sc = score + (size_t)b * N_;

    float mx = -3.4e38f;
#pragma unroll
    for (int j = 0; j < N_/256; ++j) {
        float v = sc[tid + j*256];
        buf[tid + j*256] = v;
        mx = fmaxf(mx, v);
    }
    red[tid] = mx; __syncthreads();
    for (int s = 128; s > 0; s >>= 1) { if (tid < s) red[tid] = fmaxf(red[tid], red[tid+s]); __syncthreads(); }
    mx = red[0]; __syncthreads();

    float sum = 0.f;
#pragma unroll
    for (int j = 0; j < N_/256; ++j) sum += expf(buf[tid + j*256] - mx);
    red[tid] = sum; __syncthreads();
    for (int s = 128; s > 0; s >>= 1) { if (tid < s) red[tid] += red[tid+s]; __syncthreads(); }
    const float inv = 1.f / red[0];
    __syncthreads();

    const float g     = ctl[64 + b];
    const float gamma = ctl[128 + b];
    const float s0 = ctl[256 + b*3 + 0];
    const float s1 = ctl[256 + b*3 + 1];
    const float s2 = ctl[256 + b*3 + 2];
    const float* ht = head_tm1 + (size_t)b * N_;

#pragma unroll
    for (int j = 0; j < N_/256; ++j) {
        int i = tid + j*256;
        float content = expf(buf[i] - mx) * inv;
        buf[i] = content * g + (1.f - g) * ht[i];   // inter
    }
    __syncthreads();

    float o[N_/256];
    float osum = 0.f;
#pragma unroll
    for (int j = 0; j < N_/256; ++j) {
        int i  = tid + j*256;
        int i1 = (i + N_ - 1) & (N_ - 1);
        int i2 = (i + N_ - 2) & (N_ - 1);
        float v = s0*buf[i2] + s1*buf[i1] + s2*buf[i];
        v = powf(v, gamma);
        o[j] = v;
        osum += v;
    }
    red[tid] = osum; __syncthreads();
    for (int s = 128; s > 0; s >>= 1) { if (tid < s) red[tid] += red[tid+s]; __syncthreads(); }
    const float oinv = 1.f / red[0];
#pragma unroll
    for (int j = 0; j < N_/256; ++j)
        head_out[(size_t)b * N_ + tid + j*256] = o[j] * oinv;
}

// x[:, :256] = inp
__global__ void buildx_kernel(const float* __restrict__ inp, float* __restrict__ x)
{
    int idx = blockIdx.x * 256 + threadIdx.x;           // B*IN
    int b = idx >> 8, c = idx & 255;
    x[(size_t)b * (IN_ + M_) + c] = inp[idx];
}

// x[:, 256:320] = M_read = sum_n r_head[b,n] * Mem[b,n,:]   (128MB stream, L2-hot)
__global__ __launch_bounds__(256)
void mread_kernel(const float* __restrict__ mem, const float* __restrict__ rhead,
                  float* __restrict__ x)
{
    __shared__ float red[256];
    const int b = blockIdx.x;
    const int m = threadIdx.x & 63;
    const int sub = threadIdx.x >> 6;                   // 4 n-phases
    float acc = 0.f;
    for (int n = sub; n < N_; n += 4) {
        __builtin_prefetch(&mem[((size_t)b * N_ + n + 64) * M_ + m], 0, 0);
        acc += rhead[(size_t)b * N_ + n] * mem[((size_t)b * N_ + n) * M_ + m];
    }
    red[threadIdx.x] = acc; __syncthreads();
    if (sub == 0)
        x[(size_t)b * (IN_ + M_) + IN_ + m] = red[m] + red[64+m] + red[128+m] + red[192+m];
}

// LSTM pointwise
__global__ void lstm_kernel(const float* __restrict__ gates, const float* __restrict__ c_old,
                            float* __restrict__ h_new, float* __restrict__ c_new)
{
    int idx = blockIdx.x * 256 + threadIdx.x;           // B*H
    int b = idx >> 9, hh = idx & 511;
    const float* gr = gates + (size_t)b * G4H_;
    float ig = gr[hh], fg = gr[512 + hh], gg = gr[1024 + hh], og = gr[1536 + hh];
    float si = 1.f / (1.f + expf(-ig));
    float sf = 1.f / (1.f + expf(-fg));
    float so = 1.f / (1.f + expf(-og));
    float cn = sf * c_old[idx] + si * tanhf(gg);
    c_new[idx] = cn;
    h_new[idx] = so * tanhf(cn);
}

// In-place sigmoid of erase half of write_all (4K elems) so the 33M-element
// update pass does zero transcendentals.
__global__ void sige_kernel(float* __restrict__ wall)
{
    int idx = blockIdx.x * 256 + threadIdx.x;           // B*M
    int b = idx >> 6, m = idx & 63;
    float* p = wall + (size_t)b * 2 * M_ + m;
    *p = 1.f / (1.f + expf(-*p));
}

// M_out = Mem * (1 - w*e) + w*a    (read 128MB + write 128MB, float4)
__global__ __launch_bounds__(256)
void mupdate_kernel(const float* __restrict__ mem, const float* __restrict__ whead,
                    const float* __restrict__ wall, float* __restrict__ mout)
{
    size_t gid = (size_t)blockIdx.x * 256 + threadIdx.x;    // B*N*16 float4 groups
    int b = (int)(gid / ((size_t)N_ * 16));
    size_t rem = gid % ((size_t)N_ * 16);
    int n = (int)(rem >> 4), q = (int)(rem & 15);
    float w = whead[(size_t)b * N_ + n];
    const float* er = wall + (size_t)b * 2 * M_ + q * 4;    // sigmoided erase
    const float* ar = er + M_;                              // add
    float4 mv = ((const float4*)mem)[gid];
    float4 r;
    r.x = mv.x * (1.f - w * er[0]) + w * ar[0];
    r.y = mv.y * (1.f - w * er[1]) + w * ar[1];
    r.z = mv.z * (1.f - w * er[2]) + w * ar[2];
    r.w = mv.w * (1.f - w * er[3]) + w * ar[3];
    ((float4*)mout)[gid] = r;
}

extern "C" void kernel_launch(void* const* d_in, const int* in_sizes, int n_in,
                              void* d_out, int out_size, void* d_ws, size_t ws_size,
                              hipStream_t stream)
{
    const float* inp        = (const float*)d_in[0];
    const float* h          = (const float*)d_in[1];
    const float* c          = (const float*)d_in[2];
    const float* read_head  = (const float*)d_in[3];
    const float* write_head = (const float*)d_in[4];
    const float* memory     = (const float*)d_in[5];
    const float* W_ih = (const float*)d_in[6];
    const float* b_ih = (const float*)d_in[7];
    const float* W_hh = (const float*)d_in[8];
    const float* b_hh = (const float*)d_in[9];
    const float* rk_w = (const float*)d_in[10];
    const float* rk_b = (const float*)d_in[11];
    const float* rc_w = (const float*)d_in[12];
    const float* rc_b = (const float*)d_in[13];
    const float* rs_w = (const float*)d_in[14];
    const float* rs_b = (const float*)d_in[15];
    const float* wk_w = (const float*)d_in[16];
    const float* wk_b = (const float*)d_in[17];
    const float* wc_w = (const float*)d_in[18];
    const float* wc_b = (const float*)d_in[19];
    const float* ws_w = (const float*)d_in[20];
    const float* ws_b = (const float*)d_in[21];
    const float* w_w  = (const float*)d_in[22];
    const float* w_b  = (const float*)d_in[23];

    float* out = (float*)d_out;
    float* ws  = (float*)d_ws;

    // d_out layout: h_new, c_new, r_head, w_head, M_out (flat, return order)
    float* h_new  = out;
    float* c_new  = out + (size_t)B_ * H_;
    float* r_head = out + (size_t)2 * B_ * H_;
    float* w_head = r_head + (size_t)B_ * N_;
    float* m_out  = w_head + (size_t)B_ * N_;

    float* kR    = ws + WS_KR;
    float* kW    = ws + WS_KW;
    float* ctlR  = ws + WS_CTLR;
    float* ctlW  = ws + WS_CTLW;
    float* score = ws + WS_SCORE;
    float* x     = ws + WS_X;
    float* gates = ws + WS_GATES;
    float* wall  = ws + WS_WALL;

    // ---- read phase ----
    wmma_gemm_kernel<<<(B_/16)*(M_/16), 32, 0, stream>>>(h, rk_w, rk_b, H_,
        nullptr, nullptr, nullptr, 0, kR, M_, /*tanh*/1);
    ctl_kernel<<<1, 64, 0, stream>>>(h, rc_w, rc_b, rs_w, rs_b, kR, ctlR);
    score_kernel<<<B_*(N_/256), 256, 0, stream>>>(memory, kR, ctlR, score);
    addr_kernel<<<B_, 256, 0, stream>>>(score, ctlR, read_head, r_head);

    // ---- LSTM ----
    buildx_kernel<<<B_*IN_/256, 256, 0, stream>>>(inp, x);
    mread_kernel<<<B_, 256, 0, stream>>>(memory, r_head, x);
    wmma_gemm_kernel<<<(B_/16)*(G4H_/16), 32, 0, stream>>>(x, W_ih, b_ih, IN_+M_,
        h, W_hh, b_hh, H_, gates, G4H_, /*none*/0);
    lstm_kernel<<<B_*H_/256, 256, 0, stream>>>(gates, c, h_new, c_new);

    // ---- write phase ----
    wmma_gemm_kernel<<<(B_/16)*(M_/16), 32, 0, stream>>>(h_new, wk_w, wk_b, H_,
        nullptr, nullptr, nullptr, 0, kW, M_, /*tanh*/1);
    ctl_kernel<<<1, 64, 0, stream>>>(h_new, wc_w, wc_b, ws_w, ws_b, kW, ctlW);
    score_kernel<<<B_*(N_/256), 256, 0, stream>>>(memory, kW, ctlW, score);
    addr_kernel<<<B_, 256, 0, stream>>>(score, ctlW, write_head, w_head);

    wmma_gemm_kernel<<<(B_/16)*(2*M_/16), 32, 0, stream>>>(h_new, w_w, w_b, H_,
        nullptr, nullptr, nullptr, 0, wall, 2*M_, /*none*/0);
    sige_kernel<<<B_*M_/256, 256, 0, stream>>>(wall);
    mupdate_kernel<<<(B_*N_*M_/4)/256, 256, 0, stream>>>(memory, w_head, wall, m_out);
}